// GraphTransformerModel_26121991094719
// MI455X (gfx1250) — compile-verified
//
#include <hip/hip_runtime.h>
#include <hip/hip_bf16.h>

// ---- model constants (from reference) ----
#define N_NODES  50000
#define N_EDGES  800000
#define IN_CH    128
#define HID      32
#define HEADS    4
#define HD1      (HID * HEADS)   // 128
#define N_GRAPHS 128
#define OUT_CH   10
#define KDIM     128             // K is 128 for both layers

typedef float v2f __attribute__((ext_vector_type(2)));
typedef float v8f __attribute__((ext_vector_type(8)));

// ------------------------------------------------------------------
// float atomic max via CAS (segment max for softmax)
// ------------------------------------------------------------------
__device__ __forceinline__ void atomicMaxF(float* addr, float val) {
    unsigned int* a = reinterpret_cast<unsigned int*>(addr);
    unsigned int old = __float_as_uint(*addr);
    while (__uint_as_float(old) < val) {
        unsigned int assumed = old;
        old = atomicCAS(a, assumed, __float_as_uint(val));
        if (old == assumed) break;
    }
}

// ------------------------------------------------------------------
__global__ void fill_f32(float* __restrict__ p, float v, int n) {
    int i = blockIdx.x * blockDim.x + threadIdx.x;
    if (i < n) p[i] = v;
}

// ------------------------------------------------------------------
// Fused 4-matrix projection:  Cj[M,Nout] = A[M,K] @ Wj[K,Nout] + bj
// K == 128.  Block = 256 threads = 8 waves: one 16-wide N-tile shared by
// 8 consecutive 16-row M-tiles.  All four weight panels (4 x 128 x 16)
// staged in LDS once per block; each A fragment feeds 4 WMMAs.
// WMMA fragment layout per CDNA5 ISA 7.12.2 (V_WMMA_F32_16X16X4_F32):
//   A: lane r (0..15), half h: {A[m0+r][k0+2h], A[m0+r][k0+2h+1]}
//   B: lane r, half h: rows k0+2h / k0+2h+1, column n0+r
//   C/D: elem i -> row m0+i+8h, col n0+r
// ------------------------------------------------------------------
__global__ __launch_bounds__(256) void wmma_proj4(
        const float* __restrict__ A,
        const float* __restrict__ W0, const float* __restrict__ W1,
        const float* __restrict__ W2, const float* __restrict__ W3,
        const float* __restrict__ b0, const float* __restrict__ b1,
        const float* __restrict__ b2, const float* __restrict__ b3,
        float* __restrict__ C0, float* __restrict__ C1,
        float* __restrict__ C2, float* __restrict__ C3,
        int M, int Nout) {
    const int tilesN = Nout >> 4;
    const int n0     = (blockIdx.x % tilesN) << 4;
    const int mBlk   = blockIdx.x / tilesN;          // group of 8 M-tiles
    const int tid    = threadIdx.x;
    const int waveId = tid >> 5;
    const int lane   = tid & 31;
    const int r      = lane & 15;
    const int half   = lane >> 4;

    // ---- stage the 4 weight panels in LDS (pad 16->17 to avoid conflicts)
    __shared__ float ldsB[4][KDIM][17];              // ~34 KB
#pragma unroll
    for (int i = 0; i < 8; ++i) {
        int flat = i * 256 + tid;                    // 0..2047
        int kk = flat >> 4, nn = flat & 15;
        ldsB[0][kk][nn] = W0[(size_t)kk * Nout + n0 + nn];
        ldsB[1][kk][nn] = W1[(size_t)kk * Nout + n0 + nn];
        ldsB[2][kk][nn] = W2[(size_t)kk * Nout + n0 + nn];
        ldsB[3][kk][nn] = W3[(size_t)kk * Nout + n0 + nn];
    }
    __syncthreads();

    const int m0 = (mBlk * 8 + waveId) << 4;
    if (m0 >= M) return;                             // wave-uniform

    const float* arow = A + (size_t)(m0 + r) * KDIM + 2 * half;

    v8f acc0 = {}, acc1 = {}, acc2 = {}, acc3 = {};
    for (int k0 = 0; k0 < KDIM; k0 += 4) {
        __builtin_prefetch(arow + k0 + 32, 0, 3);
        v2f a = *(const v2f*)(arow + k0);            // 8B aligned
        const int kr = k0 + 2 * half;
        v2f b;
        b.x = ldsB[0][kr][r]; b.y = ldsB[0][kr + 1][r];
        acc0 = __builtin_amdgcn_wmma_f32_16x16x4_f32(false, a, false, b, (short)0, acc0, false, false);
        b.x = ldsB[1][kr][r]; b.y = ldsB[1][kr + 1][r];
        acc1 = __builtin_amdgcn_wmma_f32_16x16x4_f32(false, a, false, b, (short)0, acc1, false, false);
        b.x = ldsB[2][kr][r]; b.y = ldsB[2][kr + 1][r];
        acc2 = __builtin_amdgcn_wmma_f32_16x16x4_f32(false, a, false, b, (short)0, acc2, false, false);
        b.x = ldsB[3][kr][r]; b.y = ldsB[3][kr + 1][r];
        acc3 = __builtin_amdgcn_wmma_f32_16x16x4_f32(false, a, false, b, (short)0, acc3, false, false);
    }

    const float bi0 = b0[n0 + r], bi1 = b1[n0 + r];
    const float bi2 = b2[n0 + r], bi3 = b3[n0 + r];
    const size_t base = (size_t)(m0 + 8 * half) * Nout + n0 + r;
#pragma unroll
    for (int i = 0; i < 8; ++i) {
        C0[base + (size_t)i * Nout] = acc0[i] + bi0;
        C1[base + (size_t)i * Nout] = acc1[i] + bi1;
        C2[base + (size_t)i * Nout] = acc2[i] + bi2;
        C3[base + (size_t)i * Nout] = acc3[i] + bi3;
    }
}

// ------------------------------------------------------------------
// per-edge attention logits + segment max (cooperative, coalesced):
// HD/4 lanes per edge, each lane one float4 of q[dst]/k[src]; 8-lane
// shfl_xor tree reduces each head (D==32 -> 8 lanes per head).
// ------------------------------------------------------------------
__global__ void edge_logits(const float* __restrict__ q, const float* __restrict__ k,
                            const int* __restrict__ src, const int* __restrict__ dst,
                            float* __restrict__ alpha, float* __restrict__ m,
                            int E, int H, int D, float scale) {
    const int HD  = H * D;
    const int lpe = HD >> 2;                         // 32 (L1) or 8 (L2)
    long long t = (long long)blockIdx.x * blockDim.x + threadIdx.x;
    long long e = t / lpe;
    int sub = (int)(t % lpe);
    if (e >= E) return;
    int s = src[e], d = dst[e];
    int f = sub << 2;
    int h = f / D;
    float4 qv = *(const float4*)(q + (size_t)d * HD + f);
    float4 kv = *(const float4*)(k + (size_t)s * HD + f);
    float p = qv.x * kv.x + qv.y * kv.y + qv.z * kv.z + qv.w * kv.w;
    p += __shfl_xor(p, 1, 32);
    p += __shfl_xor(p, 2, 32);
    p += __shfl_xor(p, 4, 32);                       // head sum (D/4 == 8 lanes)
    if ((sub & 7) == 0) {
        float logit = p * scale;
        alpha[(size_t)e * H + h] = logit;
        atomicMaxF(m + (size_t)d * H + h, logit);
    }
}

// ------------------------------------------------------------------
// ex = exp(alpha - m[dst]); denom[dst] += ex   (in-place into alpha)
// ------------------------------------------------------------------
__global__ void edge_exp(float* __restrict__ alpha, const float* __restrict__ m,
                         const int* __restrict__ dst, float* __restrict__ denom,
                         int E, int H) {
    int idx = blockIdx.x * blockDim.x + threadIdx.x;
    if (idx >= E * H) return;
    int e = idx / H, h = idx % H;
    int d = dst[e];
    float ex = __expf(alpha[idx] - m[(size_t)d * H + h]);
    alpha[idx] = ex;
    atomicAdd(denom + (size_t)d * H + h, ex);
}

// ------------------------------------------------------------------
// out[dst, :] += a(e,h) * v[src, :]   (a = ex/denom); 4 features/thread
// ------------------------------------------------------------------
__global__ void edge_scatter(const float* __restrict__ ex, const float* __restrict__ denom,
                             const float* __restrict__ v, const int* __restrict__ src,
                             const int* __restrict__ dst, float* __restrict__ out,
                             int E, int H, int D) {
    const int HD = H * D;
    const int chunks = HD >> 2;
    long long idx = (long long)blockIdx.x * blockDim.x + threadIdx.x;
    if (idx >= (long long)E * chunks) return;
    int e = (int)(idx / chunks);
    int f = (int)(idx % chunks) << 2;
    int h = f / D;
    int s = src[e], d = dst[e];
    float a = ex[(size_t)e * H + h] / (denom[(size_t)d * H + h] + 1e-16f);
    float4 vv = *(const float4*)(v + (size_t)s * HD + f);
    float* o = out + (size_t)d * HD + f;
    atomicAdd(o + 0, vv.x * a);
    atomicAdd(o + 1, vv.y * a);
    atomicAdd(o + 2, vv.z * a);
    atomicAdd(o + 3, vv.w * a);
}

// ------------------------------------------------------------------
__global__ void relu_inplace(float* __restrict__ p, int n) {
    int i = blockIdx.x * blockDim.x + threadIdx.x;
    if (i < n) p[i] = fmaxf(p[i], 0.f);
}

// ------------------------------------------------------------------
__global__ void pool_kernel(const float* __restrict__ h, const int* __restrict__ batch,
                            float* __restrict__ sums, float* __restrict__ cnt,
                            int n, int F) {
    int idx = blockIdx.x * blockDim.x + threadIdx.x;
    if (idx >= n * F) return;
    int node = idx / F, f = idx % F;
    int g = batch[node];
    atomicAdd(&sums[(size_t)g * F + f], h[idx]);
    if (f == 0) atomicAdd(&cnt[g], 1.0f);
}

// ------------------------------------------------------------------
__global__ void final_linear(const float* __restrict__ sums, const float* __restrict__ cnt,
                             const float* __restrict__ Wlin, const float* __restrict__ blin,
                             float* __restrict__ out, int G, int F, int O) {
    int idx = blockIdx.x * blockDim.x + threadIdx.x;
    if (idx >= G * O) return;
    int g = idx / O, o = idx % O;
    float inv = 1.0f / fmaxf(cnt[g], 1.0f);
    float acc = blin[o];
    for (int f = 0; f < F; ++f)
        acc += sums[(size_t)g * F + f] * inv * Wlin[(size_t)f * O + o];
    out[idx] = acc;
}

// ==================================================================
extern "C" void kernel_launch(void* const* d_in, const int* in_sizes, int n_in,
                              void* d_out, int out_size, void* d_ws, size_t ws_size,
                              hipStream_t stream) {
    const float* x    = (const float*)d_in[0];
    const int*   eidx = (const int*)  d_in[1];
    const int*   batch= (const int*)  d_in[2];
    const float* Wq1  = (const float*)d_in[3];  const float* bq1 = (const float*)d_in[4];
    const float* Wk1  = (const float*)d_in[5];  const float* bk1 = (const float*)d_in[6];
    const float* Wv1  = (const float*)d_in[7];  const float* bv1 = (const float*)d_in[8];
    const float* Ws1  = (const float*)d_in[9];  const float* bs1 = (const float*)d_in[10];
    const float* Wq2  = (const float*)d_in[11]; const float* bq2 = (const float*)d_in[12];
    const float* Wk2  = (const float*)d_in[13]; const float* bk2 = (const float*)d_in[14];
    const float* Wv2  = (const float*)d_in[15]; const float* bv2 = (const float*)d_in[16];
    const float* Ws2  = (const float*)d_in[17]; const float* bs2 = (const float*)d_in[18];
    const float* Wlin = (const float*)d_in[19]; const float* blin= (const float*)d_in[20];
    const int* src = eidx;            // edge_index[0,:]
    const int* dst = eidx + N_EDGES;  // edge_index[1,:]
    float* out = (float*)d_out;

    // ---- workspace carve (floats) ----
    float* ws = (float*)d_ws;
    size_t off = 0;
    float* bufQ  = ws + off; off += (size_t)N_NODES * HD1;
    float* bufK  = ws + off; off += (size_t)N_NODES * HD1;
    float* bufV  = ws + off; off += (size_t)N_NODES * HD1;
    float* bufH1 = ws + off; off += (size_t)N_NODES * HD1;   // skip + attn accum -> h1
    float* bufH2 = ws + off; off += (size_t)N_NODES * HID;   // skip + attn accum -> h2
    float* alpha = ws + off; off += (size_t)N_EDGES * HEADS; // logits / ex
    float* mbuf  = ws + off; off += (size_t)N_NODES * HEADS; // segment max
    float* dbuf  = ws + off; off += (size_t)N_NODES * HEADS; // segment denom
    float* sums  = ws + off; off += (size_t)N_GRAPHS * HID;
    float* cnt   = ws + off; off += (size_t)N_GRAPHS;

    const int T = 256;
    auto nb = [](long long work) { return (unsigned)((work + 255) / 256); };
    const float scale = 0.17677669529663688f;  // 1/sqrt(32)
    const int mGroups = (N_NODES / 16 + 7) / 8;        // 391 groups of 8 M-tiles

    // ================= layer 1 (H=4, D=32) =================
    wmma_proj4<<<mGroups * (HD1 / 16), T, 0, stream>>>(
        x, Wq1, Wk1, Wv1, Ws1, bq1, bk1, bv1, bs1,
        bufQ, bufK, bufV, bufH1, N_NODES, HD1);

    fill_f32<<<nb((long long)N_NODES * HEADS), T, 0, stream>>>(mbuf, -3.4e38f, N_NODES * HEADS);
    fill_f32<<<nb((long long)N_NODES * HEADS), T, 0, stream>>>(dbuf, 0.f,      N_NODES * HEADS);

    edge_logits<<<nb((long long)N_EDGES * (HD1 / 4)), T, 0, stream>>>(
        bufQ, bufK, src, dst, alpha, mbuf, N_EDGES, HEADS, HID, scale);
    edge_exp<<<nb((long long)N_EDGES * HEADS), T, 0, stream>>>(
        alpha, mbuf, dst, dbuf, N_EDGES, HEADS);
    edge_scatter<<<nb((long long)N_EDGES * (HD1 / 4)), T, 0, stream>>>(
        alpha, dbuf, bufV, src, dst, bufH1, N_EDGES, HEADS, HID);
    relu_inplace<<<nb((long long)N_NODES * HD1), T, 0, stream>>>(bufH1, N_NODES * HD1);

    // ================= layer 2 (H=1, D=32), input bufH1 =================
    wmma_proj4<<<mGroups * (HID / 16), T, 0, stream>>>(
        bufH1, Wq2, Wk2, Wv2, Ws2, bq2, bk2, bv2, bs2,
        bufQ, bufK, bufV, bufH2, N_NODES, HID);

    fill_f32<<<nb((long long)N_NODES), T, 0, stream>>>(mbuf, -3.4e38f, N_NODES);
    fill_f32<<<nb((long long)N_NODES), T, 0, stream>>>(dbuf, 0.f,      N_NODES);

    edge_logits<<<nb((long long)N_EDGES * (HID / 4)), T, 0, stream>>>(
        bufQ, bufK, src, dst, alpha, mbuf, N_EDGES, 1, HID, scale);
    edge_exp<<<nb((long long)N_EDGES), T, 0, stream>>>(
        alpha, mbuf, dst, dbuf, N_EDGES, 1);
    edge_scatter<<<nb((long long)N_EDGES * (HID / 4)), T, 0, stream>>>(
        alpha, dbuf, bufV, src, dst, bufH2, N_EDGES, 1, HID);
    relu_inplace<<<nb((long long)N_NODES * HID), T, 0, stream>>>(bufH2, N_NODES * HID);

    // ================= pool + classifier =================
    fill_f32<<<nb((long long)N_GRAPHS * HID), T, 0, stream>>>(sums, 0.f, N_GRAPHS * HID);
    fill_f32<<<1, T, 0, stream>>>(cnt, 0.f, N_GRAPHS);
    pool_kernel<<<nb((long long)N_NODES * HID), T, 0, stream>>>(
        bufH2, batch, sums, cnt, N_NODES, HID);
    final_linear<<<nb((long long)N_GRAPHS * OUT_CH), T, 0, stream>>>(
        sums, cnt, Wlin, blin, out, N_GRAPHS, HID, OUT_CH);
}